// SegmentedPolynomialNaive_85779086836285
// MI455X (gfx1250) — compile-verified
//
#include <hip/hip_runtime.h>
#include <hip/hip_bf16.h>
#include <stdint.h>
#include <stddef.h>

// ---------------------------------------------------------------------------
// Segmented tensor product on graph edges, MI455X (gfx1250, wave32).
//
// out[n] = TP( sum_{e: dst=n} x[src_e],  y[n] )   (bilinearity in x)
//
// Pipeline (all on `stream`, graph-capture safe):
//   1) kZero / kHist / kLocalScan / kScanSums / kFixup / kInitCursor / kFill
//      build a CSR of edge sources grouped by destination (in d_ws).
//   2) kMain: one wave per node. lane = channel u. Register-accumulate
//      X = sum of gathered x rows, then compute the 81x9 bilinear form with
//      V_WMMA_F32_16X16X4_F32 (M=u tiles, N=k, K=z-index chunks of 4),
//      W staged in LDS per block. Partner-half X/Y exchanged ONCE so the
//      21-chunk WMMA loop is pure VALU + ds_load_2addr(B) + wmma.
// ---------------------------------------------------------------------------

#define TOTAL   288
#define XDIM    9      // per-channel flattened feature dim (1+3+5)
#define NPATH   15
#define KTOT    84     // 81 z-indices padded to 21 chunks of 4
#define NCHUNK  21
#define WN      16     // padded N (k) dimension

typedef float v2f __attribute__((ext_vector_type(2)));
typedef float v8f __attribute__((ext_vector_type(8)));

struct CoeffPtrs { const float* p[NPATH]; };

// path segment triples (si, sj, sk); dims are 2s+1, flat offset is s*s
__device__ __constant__ int PSI[NPATH] = {0,0,1,1,1,1,0,2,1,1,2,2,2,2,2};
__device__ __constant__ int PSJ[NPATH] = {0,1,0,1,1,1,2,0,2,2,1,1,2,2,2};
__device__ __constant__ int PSK[NPATH] = {0,1,1,0,1,2,2,2,1,2,1,2,0,1,2};

// ----------------------------- CSR construction ----------------------------

__global__ void kZero(uint32_t* __restrict__ cursor, uint32_t* __restrict__ row_start, int n) {
    int i = blockIdx.x * 256 + threadIdx.x;
    if (i < n) cursor[i] = 0u;
    if (i == 0) row_start[0] = 0u;
}

__global__ void kHist(const int* __restrict__ dst, uint32_t* __restrict__ cursor, int ne) {
    int e = blockIdx.x * 256 + threadIdx.x;
    if (e < ne) atomicAdd(&cursor[dst[e]], 1u);
}

__global__ __launch_bounds__(1024) void kLocalScan(const uint32_t* __restrict__ cnt,
                                                   uint32_t* __restrict__ row_start,
                                                   uint32_t* __restrict__ bsums, int n) {
    __shared__ uint32_t s[1024];
    int t = threadIdx.x;
    int i = blockIdx.x * 1024 + t;
    s[t] = (i < n) ? cnt[i] : 0u;
    __syncthreads();
    #pragma unroll
    for (int off = 1; off < 1024; off <<= 1) {
        uint32_t add = (t >= off) ? s[t - off] : 0u;
        __syncthreads();
        s[t] += add;
        __syncthreads();
    }
    if (i < n) row_start[i + 1] = s[t];          // inclusive within block
    if (t == 1023) bsums[blockIdx.x] = s[1023];  // block total
}

__global__ __launch_bounds__(1024) void kScanSums(uint32_t* __restrict__ bsums, int nb) {
    __shared__ uint32_t s[1024];
    int t = threadIdx.x;
    s[t] = (t < nb) ? bsums[t] : 0u;
    __syncthreads();
    #pragma unroll
    for (int off = 1; off < 1024; off <<= 1) {
        uint32_t add = (t >= off) ? s[t - off] : 0u;
        __syncthreads();
        s[t] += add;
        __syncthreads();
    }
    if (t < nb) bsums[t] = (t == 0) ? 0u : s[t - 1];   // exclusive
}

__global__ void kFixup(uint32_t* __restrict__ row_start, const uint32_t* __restrict__ bsums, int n) {
    int i = blockIdx.x * 256 + threadIdx.x;
    if (i < n) row_start[i + 1] += bsums[i >> 10];
}

__global__ void kInitCursor(const uint32_t* __restrict__ row_start, uint32_t* __restrict__ cursor, int n) {
    int i = blockIdx.x * 256 + threadIdx.x;
    if (i < n) cursor[i] = row_start[i];
}

__global__ void kFill(const int* __restrict__ srcs, const int* __restrict__ dsts,
                      uint32_t* __restrict__ cursor, int* __restrict__ esrc, int ne) {
    int e = blockIdx.x * 256 + threadIdx.x;
    if (e < ne) {
        int d = dsts[e];
        uint32_t slot = atomicAdd(&cursor[d], 1u);
        esrc[slot] = srcs[e];
    }
}

// ------------------------------- main kernel -------------------------------

// z-product for compile-time flat index zz (a = zz/9, b = zz%9), 0 if padded
#define ZPROD(U, V, zz) (((zz) < 81) ? (U)[(zz) / 9] * (V)[(zz) % 9] : 0.f)

__global__ __launch_bounds__(256) void kMain(const float* __restrict__ x,
                                             const float* __restrict__ y,
                                             CoeffPtrs cf,
                                             const uint32_t* __restrict__ row_start,
                                             const int* __restrict__ esrc,
                                             float* __restrict__ out, int nnodes) {
    // W[zz=a*9+b][k] : 81x9 bilinear matrix (zero-padded to 84x16), built per block
    __shared__ float Wl[KTOT * WN];
    const int tid = threadIdx.x;
    for (int t = tid; t < KTOT * WN; t += 256) Wl[t] = 0.f;
    __syncthreads();
    #pragma unroll
    for (int p = 0; p < NPATH; ++p) {
        const int si = PSI[p], sj = PSJ[p], sk = PSK[p];
        const int di = 2 * si + 1, dj = 2 * sj + 1, dk = 2 * sk + 1;
        const int nel = di * dj * dk;
        const float* cp = cf.p[p];
        for (int t = tid; t < nel; t += 256) {
            int i  = t / (dj * dk);
            int r  = t - i * dj * dk;
            int j  = r / dk;
            int kk = r - j * dk;
            int a = si * si + i, b = sj * sj + j, k = sk * sk + kk;
            Wl[(a * XDIM + b) * WN + k] = cp[t];   // paths are disjoint triples
        }
    }
    __syncthreads();

    const int wave = tid >> 5;
    const int lane = tid & 31;
    const int node = blockIdx.x * 8 + wave;
    if (node >= nnodes) return;                    // wave-uniform exit

    // ---- aggregate X = sum over incoming edges of x[src], channel = lane ----
    float X[XDIM], Y[XDIM];
    #pragma unroll
    for (int a = 0; a < XDIM; ++a) X[a] = 0.f;
    const uint32_t beg = (uint32_t)__builtin_amdgcn_readfirstlane((int)row_start[node]);
    const uint32_t end = (uint32_t)__builtin_amdgcn_readfirstlane((int)row_start[node + 1]);
    for (uint32_t t = beg; t < end; ++t) {
        const int src = __builtin_amdgcn_readfirstlane(esrc[t]);
        const float* xb = x + (size_t)src * TOTAL;
        X[0] += xb[lane];
        const float* x1 = xb + 32 + 3 * lane;      // -> global_load_b96
        X[1] += x1[0]; X[2] += x1[1]; X[3] += x1[2];
        const float* x2 = xb + 128 + 5 * lane;     // -> b128 + b32
        X[4] += x2[0]; X[5] += x2[1]; X[6] += x2[2]; X[7] += x2[3]; X[8] += x2[4];
    }
    {
        const float* yb = y + (size_t)node * TOTAL;
        Y[0] = yb[lane];
        const float* y1 = yb + 32 + 3 * lane;
        Y[1] = y1[0]; Y[2] = y1[1]; Y[3] = y1[2];
        const float* y2 = yb + 128 + 5 * lane;
        Y[4] = y2[0]; Y[5] = y2[1]; Y[6] = y2[2]; Y[7] = y2[3]; Y[8] = y2[4];
    }

#if __has_builtin(__builtin_amdgcn_wmma_f32_16x16x4_f32)
    // ---- D(16x16) = A(16x4) * B(4x16) + C, looped over 21 K-chunks ----
    // A layout (ISA 7.12.2): lanes 0-15 hold M=lane with K=0,1 in VGPR0,1;
    //                        lanes 16-31 hold M=lane-16 with K=2,3.
    // B layout (row striped across lanes): VGPR0 = rows K=0 (lanes 0-15) /
    //                        K=2 (lanes 16-31); VGPR1 = rows K=1 / K=3.
    const bool lo = (lane < 16);
    const int  lc = lane & 15;

    // One-time half-wave exchange of the feature vectors (18 permutes total),
    // then build per-lane operand sets:
    //   U0/V0 = features of channel lc        (rows of tile 0)
    //   U1/V1 = features of channel lc + 16   (rows of tile 1)
    float Xp[XDIM], Yp[XDIM];
    #pragma unroll
    for (int a = 0; a < XDIM; ++a) {
        Xp[a] = __shfl_xor(X[a], 16, 32);
        Yp[a] = __shfl_xor(Y[a], 16, 32);
    }
    float U0[XDIM], V0[XDIM], U1[XDIM], V1[XDIM];
    #pragma unroll
    for (int a = 0; a < XDIM; ++a) {
        U0[a] = lo ? X[a]  : Xp[a];
        V0[a] = lo ? Y[a]  : Yp[a];
        U1[a] = lo ? Xp[a] : X[a];
        V1[a] = lo ? Yp[a] : Y[a];
    }

    const float* Bp = &Wl[(lo ? 0 : 2) * WN + lc];

    v8f acc0 = {0.f,0.f,0.f,0.f,0.f,0.f,0.f,0.f};  // channels 0..15
    v8f acc1 = {0.f,0.f,0.f,0.f,0.f,0.f,0.f,0.f};  // channels 16..31

    #pragma unroll
    for (int c = 0; c < NCHUNK; ++c) {
        // lanes 0-15 supply K = 4c,4c+1 ; lanes 16-31 supply K = 4c+2,4c+3
        const int zL0 = 4 * c, zL1 = 4 * c + 1, zH0 = 4 * c + 2, zH1 = 4 * c + 3;
        v2f A0, A1, B;
        A0.x = lo ? ZPROD(U0, V0, zL0) : ZPROD(U0, V0, zH0);
        A0.y = lo ? ZPROD(U0, V0, zL1) : ZPROD(U0, V0, zH1);
        A1.x = lo ? ZPROD(U1, V1, zL0) : ZPROD(U1, V1, zH0);
        A1.y = lo ? ZPROD(U1, V1, zL1) : ZPROD(U1, V1, zH1);
        B.x = Bp[c * 64];       B.y = Bp[c * 64 + 16];   // W rows 4c+{0..3}
        acc0 = __builtin_amdgcn_wmma_f32_16x16x4_f32(false, A0, false, B,
                                                     (short)0, acc0, false, false);
        acc1 = __builtin_amdgcn_wmma_f32_16x16x4_f32(false, A1, false, B,
                                                     (short)0, acc1, false, false);
    }

    // D layout: lane lc = column k; lanes 0-15 hold rows 0..7, lanes 16-31 rows 8..15
    if (lc < XDIM) {
        int segoff, d, i;
        if      (lc == 0) { segoff = 0;   d = 1; i = 0;      }
        else if (lc <  4) { segoff = 32;  d = 3; i = lc - 1; }
        else              { segoff = 128; d = 5; i = lc - 4; }
        float* ob = out + (size_t)node * TOTAL + segoff + i;
        const int u0 = lo ? 0 : 8;
        #pragma unroll
        for (int r = 0; r < 8; ++r) {
            ob[(size_t)(u0 + r) * d]        = acc0[r];
            ob[(size_t)(u0 + r + 16) * d]   = acc1[r];
        }
    }
#else
    // Fallback: per-lane VALU contraction out[k] = sum_zz z[zz]*W[zz][k]
    float o[XDIM];
    #pragma unroll
    for (int k = 0; k < XDIM; ++k) o[k] = 0.f;
    #pragma unroll
    for (int a = 0; a < XDIM; ++a)
        #pragma unroll
        for (int b = 0; b < XDIM; ++b) {
            const float zv = X[a] * Y[b];
            const float* wr = &Wl[(a * XDIM + b) * WN];
            #pragma unroll
            for (int k = 0; k < XDIM; ++k) o[k] += zv * wr[k];
        }
    float* ob = out + (size_t)node * TOTAL;
    ob[lane] = o[0];
    float* o1 = ob + 32 + 3 * lane;
    o1[0] = o[1]; o1[1] = o[2]; o1[2] = o[3];
    float* o2 = ob + 128 + 5 * lane;
    o2[0] = o[4]; o2[1] = o[5]; o2[2] = o[6]; o2[3] = o[7]; o2[4] = o[8];
#endif
}

// ------------------------------- launcher ----------------------------------

extern "C" void kernel_launch(void* const* d_in, const int* in_sizes, int n_in,
                              void* d_out, int out_size, void* d_ws, size_t ws_size,
                              hipStream_t stream) {
    static const int psz[NPATH] = {1,9,9,9,27,45,25,25,45,75,45,75,25,75,125};

    const float* x = (const float*)d_in[0];
    const float* y = (const float*)d_in[1];
    CoeffPtrs cf;
    const int* idx_src;
    const int* idx_dst;
    int ne;
    if (n_in == 5) {
        // coeffs flattened into one 615-element array
        const float* base = (const float*)d_in[2];
        size_t off = 0;
        for (int p = 0; p < NPATH; ++p) { cf.p[p] = base + off; off += (size_t)psz[p]; }
        idx_src = (const int*)d_in[3];
        idx_dst = (const int*)d_in[4];
        ne = in_sizes[3];
    } else {
        // coeff tuple flattened into 15 separate inputs: d_in[2..16]
        for (int p = 0; p < NPATH; ++p) cf.p[p] = (const float*)d_in[2 + p];
        idx_src = (const int*)d_in[n_in - 2];
        idx_dst = (const int*)d_in[n_in - 1];
        ne = in_sizes[n_in - 2];
    }
    const int nnodes = in_sizes[0] / TOTAL;

    // workspace layout: row_start | cursor | esrc | bsums  (~2.5 MB)
    uint32_t* row_start = (uint32_t*)d_ws;
    uint32_t* cursor    = row_start + (nnodes + 1);
    int*      esrc      = (int*)(cursor + nnodes);
    uint32_t* bsums     = (uint32_t*)(esrc + ne);
    (void)ws_size; (void)out_size;

    const int nb = (nnodes + 1023) / 1024;    // scan blocks (<= 1024 assumed)

    kZero      <<<(nnodes + 255) / 256, 256, 0, stream>>>(cursor, row_start, nnodes);
    kHist      <<<(ne + 255) / 256,     256, 0, stream>>>(idx_dst, cursor, ne);
    kLocalScan <<<nb,                  1024, 0, stream>>>(cursor, row_start, bsums, nnodes);
    kScanSums  <<<1,                   1024, 0, stream>>>(bsums, nb);
    kFixup     <<<(nnodes + 255) / 256, 256, 0, stream>>>(row_start, bsums, nnodes);
    kInitCursor<<<(nnodes + 255) / 256, 256, 0, stream>>>(row_start, cursor, nnodes);
    kFill      <<<(ne + 255) / 256,     256, 0, stream>>>(idx_src, idx_dst, cursor, esrc, ne);
    kMain      <<<(nnodes + 7) / 8,     256, 0, stream>>>(x, y, cf, row_start, esrc,
                                                          (float*)d_out, nnodes);
}